// Model_6158983102572
// MI455X (gfx1250) — compile-verified
//
#include <hip/hip_runtime.h>

// ---------------------------------------------------------------------------
// MI455X (gfx1250) fused GNN pipeline.
//   B=2, N=10000, E=160000, IDIM=10, HDIM=32, IWIN=48, HWIN=16, OWIN=24, ODIM=1
// Bandwidth-bound problem (~2.4 GFLOP vs ~400MB naive traffic): fuse stage1+2
// and stage4+5 per node; graph conv via device-built CSR gather (no f32 atomics).
// Matrix math uses V_WMMA_F32_16X16X4_F32 (exact f32, wave32, 16x16 tiles).
// ---------------------------------------------------------------------------

typedef __attribute__((ext_vector_type(2))) float v2f;
typedef __attribute__((ext_vector_type(8))) float v8f;

static constexpr int NB    = 2;       // batch
static constexpr int NN    = 10000;   // nodes
static constexpr int NE    = 160000;  // edges
static constexpr int IDIM  = 10;
static constexpr int HDIM  = 32;
static constexpr int IWIN  = 48;
static constexpr int HWIN  = 16;
static constexpr int OWIN  = 24;
static constexpr int TILE  = HWIN * HDIM;   // 512 floats per (b,node) h2/x3 tile

// ---------------------------------------------------------------------------
// Stage 1+2 fused: h2 = relu(W_tin @ relu(feat @ W_fin^T + b_fin) + b_tin)
// One wave per (b,n). 42 wmma_f32_16x16x4 per wave.
// LDS per wave: sF 48x12 (576 f, zero-padded K) + sH1 48x32 (1536 f); h2
// staging overlays sF. 4 waves/block -> 33,792B static LDS.
// ---------------------------------------------------------------------------
__global__ __launch_bounds__(128) void k_stage12(
    const float* __restrict__ feat, const float* __restrict__ Wfin,
    const float* __restrict__ bfin, const float* __restrict__ Wtin,
    const float* __restrict__ btin, float* __restrict__ h2out) {
  __shared__ float smem[4 * 2112];
  const int lane = threadIdx.x & 31;
  const int wib  = threadIdx.x >> 5;
  const int idx  = blockIdx.x * 4 + wib;      // 0 .. B*N-1, exact
  float* sF  = &smem[wib * 2112];             // 48 x 12 (cols 10,11 zero)
  float* sH1 = sF + 576;                      // 48 x 32
  float* sH2 = sF;                            // 16 x 32, overlays sF later

  const int half = (lane < 16) ? 0 : 2;       // f32 WMMA A/B K-split per lane half
  const int l16  = lane & 15;

  // ---- load feat tile (48x10, coalesced) into padded 48x12 LDS ----
  const float* fsrc = feat + (size_t)idx * (IWIN * IDIM);
  for (int i = lane; i < IWIN * IDIM; i += 32) {
    int t = i / IDIM, f = i - t * IDIM;
    sF[t * 12 + f] = fsrc[i];
  }
  for (int i = lane; i < IWIN * 2; i += 32) {        // zero K-pad cols 10..11
    int t = i >> 1;
    sF[t * 12 + 10 + (i & 1)] = 0.f;
  }

  // ---- constant B fragments for GEMM1: B[k][h] = W_fin[h][k], K padded to 12
  v2f b1[3][2];
  for (int ks = 0; ks < 3; ++ks)
    for (int nt = 0; nt < 2; ++nt) {
      int h  = nt * 16 + l16;
      int k0 = ks * 4 + half;
      b1[ks][nt].x = (k0     < IDIM) ? Wfin[h * IDIM + k0]     : 0.f;
      b1[ks][nt].y = (k0 + 1 < IDIM) ? Wfin[h * IDIM + k0 + 1] : 0.f;
    }
  // ---- constant A fragments for GEMM2: A[s][t] = W_tin[s][t], K=48 exact ----
  v2f a2[12];
  for (int ks = 0; ks < 12; ++ks) {
    int k0 = ks * 4 + half;
    a2[ks].x = Wtin[l16 * IWIN + k0];
    a2[ks].y = Wtin[l16 * IWIN + k0 + 1];
  }

  __syncthreads();

  // ---- GEMM1: h1(48x32) = relu(F(48x10) @ Wfin^T + b_fin) ----
  for (int mt = 0; mt < 3; ++mt)
    for (int nt = 0; nt < 2; ++nt) {
      v8f c;
      float bb = bfin[nt * 16 + l16];           // bias per column h
      for (int v = 0; v < 8; ++v) c[v] = bb;
      for (int ks = 0; ks < 3; ++ks) {
        v2f a;
        int row = mt * 16 + l16;
        int k0  = ks * 4 + half;
        a.x = sF[row * 12 + k0];
        a.y = sF[row * 12 + k0 + 1];
        c = __builtin_amdgcn_wmma_f32_16x16x4_f32(false, a, false, b1[ks][nt],
                                                  (short)0, c, false, false);
      }
      for (int v = 0; v < 8; ++v) {
        int row = mt * 16 + v + ((lane < 16) ? 0 : 8);
        int col = nt * 16 + l16;
        sH1[row * 32 + col] = fmaxf(c[v], 0.f);
      }
    }

  __syncthreads();

  // ---- GEMM2: h2(16x32) = relu(W_tin(16x48) @ h1 + b_tin) ----
  for (int nt = 0; nt < 2; ++nt) {
    v8f c;
    for (int v = 0; v < 8; ++v)
      c[v] = btin[v + ((lane < 16) ? 0 : 8)];   // bias per row s
    for (int ks = 0; ks < 12; ++ks) {
      v2f bf;
      int col = nt * 16 + l16;
      int k0  = ks * 4 + half;
      bf.x = sH1[k0 * 32 + col];
      bf.y = sH1[(k0 + 1) * 32 + col];
      c = __builtin_amdgcn_wmma_f32_16x16x4_f32(false, a2[ks], false, bf,
                                                (short)0, c, false, false);
    }
    for (int v = 0; v < 8; ++v) {
      int row = v + ((lane < 16) ? 0 : 8);
      int col = nt * 16 + l16;
      sH2[row * 32 + col] = fmaxf(c[v], 0.f);
    }
  }

  __syncthreads();

  // ---- coalesced float4 write of the 512-float tile ----
  float4* dst = (float4*)(h2out + (size_t)idx * TILE);
  const float4* s4 = (const float4*)sH2;
  for (int i = lane; i < TILE / 4; i += 32) dst[i] = s4[i];
}

// ---------------------------------------------------------------------------
// CSR build: zero, count, scan, fill
// ---------------------------------------------------------------------------
__global__ void k_zero(int* __restrict__ deg, int* __restrict__ cur) {
  int i = blockIdx.x * blockDim.x + threadIdx.x;
  if (i < NN) { deg[i] = 0; cur[i] = 0; }
}

__global__ void k_count(const int* __restrict__ dst, int* __restrict__ deg) {
  int e = blockIdx.x * blockDim.x + threadIdx.x;
  if (e < NE) atomicAdd(&deg[dst[e]], 1);
}

__global__ __launch_bounds__(1024) void k_scan(const int* __restrict__ deg,
                                               int* __restrict__ off) {
  __shared__ int buf[1024];
  __shared__ int carry;
  int tid = threadIdx.x;
  if (tid == 0) { carry = 0; off[0] = 0; }
  __syncthreads();
  for (int base = 0; base < NN; base += 1024) {
    int i = base + tid;
    buf[tid] = (i < NN) ? deg[i] : 0;
    __syncthreads();
    for (int s = 1; s < 1024; s <<= 1) {        // Hillis-Steele inclusive scan
      int t = (tid >= s) ? buf[tid - s] : 0;
      __syncthreads();
      buf[tid] += t;
      __syncthreads();
    }
    if (i < NN) off[i + 1] = carry + buf[tid];
    __syncthreads();
    if (tid == 0) carry += buf[1023];
    __syncthreads();
  }
}

__global__ void k_fill(const int* __restrict__ dst, const int* __restrict__ off,
                       int* __restrict__ cur, int* __restrict__ eidx) {
  int e = blockIdx.x * blockDim.x + threadIdx.x;
  if (e < NE) {
    int d = dst[e];
    int p = atomicAdd(&cur[d], 1);
    eidx[off[d] + p] = e;
  }
}

// ---------------------------------------------------------------------------
// Graph conv: x3 = relu(mean over incident edges of w[e]*h2[src[e]]).
// One wave per node; both batch slices accumulated in registers (8 float4/lane).
// Edge list sorted by edge index (lane 0) => deterministic float sum order.
// ---------------------------------------------------------------------------
__global__ __launch_bounds__(256) void k_graph(
    const float* __restrict__ h2, const float* __restrict__ w,
    const int* __restrict__ srcv, const int* __restrict__ off,
    const int* __restrict__ eidx, float* __restrict__ x3) {
  __shared__ int elist[8][256];
  const int lane = threadIdx.x & 31;
  const int wib  = threadIdx.x >> 5;
  const int n    = blockIdx.x * 8 + wib;      // 0..NN-1, exact
  const int s0 = off[n], s1 = off[n + 1];
  const int deg = s1 - s0;
  const int cnt = deg < 256 ? deg : 256;      // Poisson(16): >256 is impossible

  for (int j = lane; j < cnt; j += 32) elist[wib][j] = eidx[s0 + j];
  __syncthreads();
  if (lane == 0) {                            // insertion sort -> canonical order
    for (int i = 1; i < cnt; ++i) {
      int key = elist[wib][i], j = i - 1;
      while (j >= 0 && elist[wib][j] > key) { elist[wib][j + 1] = elist[wib][j]; --j; }
      elist[wib][j + 1] = key;
    }
  }
  __syncthreads();

  float4 acc[8];
  for (int q = 0; q < 8; ++q) { acc[q].x = acc[q].y = acc[q].z = acc[q].w = 0.f; }

  for (int j = 0; j < deg; ++j) {
    int e = (j < 256) ? elist[wib][j] : eidx[s0 + j];
    float wv = w[e];
    int sn = srcv[e];
    const float4* p0 = (const float4*)(h2 + (size_t)sn * TILE);          // b=0
    const float4* p1 = (const float4*)(h2 + (size_t)(NN + sn) * TILE);   // b=1
    for (int q = 0; q < 4; ++q) {
      float4 m0 = p0[q * 32 + lane];
      float4 m1 = p1[q * 32 + lane];
      acc[q].x += wv * m0.x; acc[q].y += wv * m0.y;
      acc[q].z += wv * m0.z; acc[q].w += wv * m0.w;
      acc[4 + q].x += wv * m1.x; acc[4 + q].y += wv * m1.y;
      acc[4 + q].z += wv * m1.z; acc[4 + q].w += wv * m1.w;
    }
  }

  const float invd = (deg > 0) ? 1.0f / (float)deg : 0.f;
  float4* o0 = (float4*)(x3 + (size_t)n * TILE);
  float4* o1 = (float4*)(x3 + (size_t)(NN + n) * TILE);
  for (int q = 0; q < 4; ++q) {
    float4 r0, r1;
    r0.x = fmaxf(acc[q].x * invd, 0.f); r0.y = fmaxf(acc[q].y * invd, 0.f);
    r0.z = fmaxf(acc[q].z * invd, 0.f); r0.w = fmaxf(acc[q].w * invd, 0.f);
    r1.x = fmaxf(acc[4 + q].x * invd, 0.f); r1.y = fmaxf(acc[4 + q].y * invd, 0.f);
    r1.z = fmaxf(acc[4 + q].z * invd, 0.f); r1.w = fmaxf(acc[4 + q].w * invd, 0.f);
    o0[q * 32 + lane] = r0;
    o1[q * 32 + lane] = r1;
  }
}

// ---------------------------------------------------------------------------
// Stage 4+5 fused: out = relu(W_tout @ x3 + b_tout) @ W_fout^T + b_fout
// One wave per (b,n). 16 wmma_f32_16x16x4 per wave (M padded 24->32).
// ---------------------------------------------------------------------------
__global__ __launch_bounds__(128) void k_stage45(
    const float* __restrict__ x3, const float* __restrict__ Wtout,
    const float* __restrict__ btout, const float* __restrict__ Wfout,
    const float* __restrict__ bfout, float* __restrict__ out) {
  __shared__ float smem[4 * 1536];
  const int lane = threadIdx.x & 31;
  const int wib  = threadIdx.x >> 5;
  const int idx  = blockIdx.x * 4 + wib;      // 0 .. B*N-1, exact
  float* sX  = &smem[wib * 1536];             // 16 x 32
  float* sH4 = sX + 512;                      // 32 x 32 (rows >=24 junk)

  const int half = (lane < 16) ? 0 : 2;
  const int l16  = lane & 15;

  const float4* src4 = (const float4*)(x3 + (size_t)idx * TILE);
  float4* sX4 = (float4*)sX;
  for (int i = lane; i < TILE / 4; i += 32) sX4[i] = src4[i];
  __syncthreads();

  for (int mt = 0; mt < 2; ++mt)
    for (int nt = 0; nt < 2; ++nt) {
      v8f c;
      for (int v = 0; v < 8; ++v) {
        int row = mt * 16 + v + ((lane < 16) ? 0 : 8);
        c[v] = (row < OWIN) ? btout[row] : 0.f;
      }
      for (int ks = 0; ks < 4; ++ks) {
        int k0 = ks * 4 + half;
        v2f a, bf;
        int row = mt * 16 + l16;
        a.x = (row < OWIN) ? Wtout[row * HWIN + k0]     : 0.f;
        a.y = (row < OWIN) ? Wtout[row * HWIN + k0 + 1] : 0.f;
        int col = nt * 16 + l16;
        bf.x = sX[k0 * 32 + col];
        bf.y = sX[(k0 + 1) * 32 + col];
        c = __builtin_amdgcn_wmma_f32_16x16x4_f32(false, a, false, bf,
                                                  (short)0, c, false, false);
      }
      for (int v = 0; v < 8; ++v) {
        int row = mt * 16 + v + ((lane < 16) ? 0 : 8);
        int col = nt * 16 + l16;
        sH4[row * 32 + col] = fmaxf(c[v], 0.f);
      }
    }
  __syncthreads();

  // Stage 5: 32->1 feature projection (tiny; VALU dot per output row)
  if (lane < OWIN) {
    float s = bfout[0];
    for (int h = 0; h < HDIM; ++h) s += sH4[lane * 32 + h] * Wfout[h];
    out[(size_t)idx * OWIN + lane] = s;
  }
}

// ---------------------------------------------------------------------------
extern "C" void kernel_launch(void* const* d_in, const int* in_sizes, int n_in,
                              void* d_out, int out_size, void* d_ws, size_t ws_size,
                              hipStream_t stream) {
  const float* feat  = (const float*)d_in[0];
  const float* w     = (const float*)d_in[1];
  const int*   src   = (const int*)d_in[2];
  const int*   dst   = (const int*)d_in[3];
  const float* Wfin  = (const float*)d_in[4];
  const float* bfin  = (const float*)d_in[5];
  const float* Wtin  = (const float*)d_in[6];
  const float* btin  = (const float*)d_in[7];
  const float* Wtout = (const float*)d_in[8];
  const float* btout = (const float*)d_in[9];
  const float* Wfout = (const float*)d_in[10];
  const float* bfout = (const float*)d_in[11];
  float* out = (float*)d_out;

  // workspace carve-up (256B aligned)
  char* base = (char*)d_ws;
  auto au = [](size_t x) { return (x + 255) & ~(size_t)255; };
  size_t o = 0;
  int* deg  = (int*)(base + o); o = au(o + (size_t)NN * 4);
  int* off  = (int*)(base + o); o = au(o + (size_t)(NN + 1) * 4);
  int* cur  = (int*)(base + o); o = au(o + (size_t)NN * 4);
  int* eidx = (int*)(base + o); o = au(o + (size_t)NE * 4);
  float* h2 = (float*)(base + o); o = au(o + (size_t)NB * NN * TILE * 4);
  float* x3 = (float*)(base + o);

  // CSR build
  k_zero <<<(NN + 255) / 256, 256, 0, stream>>>(deg, cur);
  k_count<<<(NE + 255) / 256, 256, 0, stream>>>(dst, deg);
  k_scan <<<1, 1024, 0, stream>>>(deg, off);
  k_fill <<<(NE + 255) / 256, 256, 0, stream>>>(dst, off, cur, eidx);

  // fused pipeline
  k_stage12<<<(NB * NN) / 4, 128, 0, stream>>>(feat, Wfin, bfin, Wtin, btin, h2);
  k_graph  <<<NN / 8, 256, 0, stream>>>(h2, w, src, off, eidx, x3);
  k_stage45<<<(NB * NN) / 4, 128, 0, stream>>>(x3, Wtout, btout, Wfout, bfout, out);
}